// ES_Normalize_29463475650709
// MI455X (gfx1250) — compile-verified
//
#include <hip/hip_runtime.h>

// EMA scan y[t] = a*y[t-1] + (1-a)*x[t] (y[0]=x[0]) over (32,4096,256) f32.
// Memory-bound: ~320 MiB traffic incl. warm-up => ~14 us floor @ 23.3 TB/s.
// Per 16x16 (time x channel) tile the scan is D = L(16x16 Toeplitz of alpha
// powers) @ B + alpha^{m+1}*carry, done as 4 chained V_WMMA_F32_16X16X4_F32.

typedef __attribute__((ext_vector_type(2))) float v2f;
typedef __attribute__((ext_vector_type(8))) float v8f;

namespace {
constexpr float kAlpha = 0.9f;
constexpr float kOneMinusAlpha = 1.0f - kAlpha;
constexpr int kB = 32;
constexpr int kT = 4096;
constexpr int kC = 256;
constexpr int kSeg = 8;              // T split into 8 independent segments
constexpr int kSegLen = kT / kSeg;   // 512
constexpr int kWarm = 256;           // alpha^256 ~ 2e-12 << f32 eps of result
constexpr int kCTiles = kC / 16;     // 16 channel tiles
}

__global__ __launch_bounds__(32)
void ema_scan_wmma(const float* __restrict__ x, float* __restrict__ out) {
  const int bid   = (int)blockIdx.x;
  const int ctile = bid % kCTiles;
  const int seg   = (bid / kCTiles) % kSeg;
  const int b     = bid / (kCTiles * kSeg);

  const int lane = (int)threadIdx.x;   // wave32
  const int n    = lane & 15;          // channel within tile / A row m
  const bool hi  = lane >= 16;
  const int r0   = hi ? 2 : 0;         // K rows held by this lane half

  const size_t base = (size_t)b * kT * kC + (size_t)ctile * 16 + (size_t)n;
  const float* __restrict__ xp = x + base;
  float* __restrict__ op = out + base;

  // alpha^0 .. alpha^16
  float ap[17];
  ap[0] = 1.0f;
#pragma unroll
  for (int i = 1; i <= 16; ++i) ap[i] = ap[i - 1] * kAlpha;

  // A fragments: A[m][k] = (m>=k) ? alpha^(m-k) : 0, 16x4 per WMMA step.
  // Layout (ISA 7.12.2): lanes 0-15 hold K=k0,k0+1 ; lanes 16-31 K=k0+2,k0+3.
  v2f A[4];
#pragma unroll
  for (int kk = 0; kk < 4; ++kk) {
    const int k0 = 4 * kk + r0;
    A[kk].x = (n >= k0)     ? ap[n - k0]     : 0.0f;
    A[kk].y = (n >= k0 + 1) ? ap[n - k0 - 1] : 0.0f;
  }

  const int tStore = seg * kSegLen;
  const int tBegin = (seg == 0) ? 0 : (tStore - kWarm);
  const int tEnd   = tStore + kSegLen;

  const float hiAdj = hi ? ap[8] : 1.0f;  // upper lanes hold rows m = v+8

  float cur[8], nxt[8];
  auto loadTile = [&](int t0, float dst[8]) {
#pragma unroll
    for (int kk = 0; kk < 4; ++kk) {
      const size_t t = (size_t)(t0 + 4 * kk + r0);
      dst[2 * kk + 0] = xp[t * kC];
      dst[2 * kk + 1] = xp[(t + 1) * kC];
    }
  };

  loadTile(tBegin, cur);
  float carry = 0.0f;

  for (int t0 = tBegin; t0 < tEnd; t0 += 16) {
    if (t0 + 16 < tEnd) loadTile(t0 + 16, nxt);   // double-buffer next tile

    // C[v] = alpha^{m+1} * carry[n], m = v (+8 for upper lanes)
    v8f acc;
#pragma unroll
    for (int v = 0; v < 8; ++v) acc[v] = carry * hiAdj * ap[v + 1];

#pragma unroll
    for (int kk = 0; kk < 4; ++kk) {
      const int t = t0 + 4 * kk + r0;
      v2f Bf;  // b[t] = x[t] * (1-a), except the very first sample of the seq
      Bf.x = cur[2 * kk + 0] * ((t == 0) ? 1.0f : kOneMinusAlpha);
      Bf.y = cur[2 * kk + 1] * kOneMinusAlpha;
      acc = __builtin_amdgcn_wmma_f32_16x16x4_f32(
          /*neg_a=*/false, A[kk], /*neg_b=*/false, Bf,
          /*c_mod=*/(short)0, acc, /*reuse_a=*/false, /*reuse_b=*/false);
    }

    // New carry = D[15][n]: VGPR 7 of lane (16+n)
    carry = __shfl(acc[7], 16 + n, 32);

    if (t0 >= tStore) {
      const int rowBase = t0 + (hi ? 8 : 0);
#pragma unroll
      for (int v = 0; v < 8; ++v) op[(size_t)(rowBase + v) * kC] = acc[v];
    }
#pragma unroll
    for (int i = 0; i < 8; ++i) cur[i] = nxt[i];
  }
}

// tail = out[:, T-1, 0]
__global__ void ema_tail(const float* __restrict__ out, float* __restrict__ tail) {
  const int b = (int)threadIdx.x;
  if (b < kB) tail[b] = out[(size_t)b * kT * kC + (size_t)(kT - 1) * kC];
}

extern "C" void kernel_launch(void* const* d_in, const int* in_sizes, int n_in,
                              void* d_out, int out_size, void* d_ws, size_t ws_size,
                              hipStream_t stream) {
  (void)in_sizes; (void)n_in; (void)d_ws; (void)ws_size; (void)out_size;
  const float* x = (const float*)d_in[0];
  float* out = (float*)d_out;
  float* tail = out + (size_t)kB * kT * kC;

  ema_scan_wmma<<<dim3(kB * kSeg * kCTiles), dim3(32), 0, stream>>>(x, out);
  ema_tail<<<dim3(1), dim3(kB), 0, stream>>>(out, tail);
}